// RadiusInteractionGraph_1898375545375
// MI455X (gfx1250) — compile-verified
//
#include <hip/hip_runtime.h>
#include <math.h>

// CDNA5 / gfx1250, wave32.
typedef __attribute__((ext_vector_type(2))) float v2f;
typedef __attribute__((ext_vector_type(8))) float v8f;

#define K_NEI    32
#define CUTOFF2  100.0f        // CUTOFF^2 = 10^2
#define TILE_M   16

// ---------------------------------------------------------------------------
// Kernel 1: molecule [start,end) bounds from the sorted batch vector.
// ---------------------------------------------------------------------------
__global__ void mol_bounds_kernel(const int* __restrict__ batch, int n,
                                  int* __restrict__ mstart, int* __restrict__ mend) {
    int t = blockIdx.x * blockDim.x + threadIdx.x;
    if (t >= n) return;
    int b = batch[t];
    if (t == 0     || batch[t - 1] != b) mstart[b] = t;
    if (t == n - 1 || batch[t + 1] != b) mend[b]   = t + 1;
}

// ---------------------------------------------------------------------------
// Kernel 2: one wave32 per 16-target tile. WMMA computes 16x16 d^2 tiles:
//   A = -2 * P_targets (16x4, coord 3 padded 0)   B = P_candidates (4x16)
//   C[m][n] = |p_m|^2 + |p_n|^2   =>   D = d^2 tile in one v_wmma.
// ---------------------------------------------------------------------------
__global__ __launch_bounds__(32)
void radius_graph_wmma_kernel(const float* __restrict__ pos,
                              const int*   __restrict__ batch,
                              const int*   __restrict__ mstart,
                              const int*   __restrict__ mend,
                              float*       __restrict__ out,
                              int n) {
    __shared__ float ldsNormT[TILE_M];
    __shared__ float ldsD2[TILE_M][TILE_M + 1];   // +1 pad vs bank conflicts

    const int lane = threadIdx.x;       // 0..31
    const int half = lane >> 4;         // 0: K=0,1 rows | 1: K=2,3 rows
    const int l16  = lane & 15;
    const int i    = blockIdx.x * TILE_M + l16;   // this lane's target atom
    const int isafe = min(i, n - 1);

    // Target position / norm / molecule range (contiguous since batch sorted).
    const float tx = pos[3 * isafe + 0];
    const float ty = pos[3 * isafe + 1];
    const float tz = pos[3 * isafe + 2];
    const float normT = tx * tx + ty * ty + tz * tz;
    const int   bi = batch[isafe];
    const int   si = mstart[bi];
    const int   ei = mend[bi];

    // A fragment (16x4 f32, VGPR layout: lanes 0-15 K=0/1, lanes 16-31 K=2/3),
    // pre-scaled by -2 so the WMMA emits d^2 directly.
    v2f afrag;
    if (half == 0) { afrag.x = -2.0f * tx; afrag.y = -2.0f * ty; }
    else           { afrag.x = -2.0f * tz; afrag.y =  0.0f;      }

    if (half == 0) ldsNormT[l16] = normT;
    __syncthreads();

    // Loop-invariant: the 8 target norms this lane's accumulator rows need
    // (C/D layout: lane L, VGPR v -> M = v + 8*(L/16)).
    float normT_reg[8];
#pragma unroll
    for (int v = 0; v < 8; ++v) normT_reg[v] = ldsNormT[v + 8 * half];

    // Candidate window = union of molecule ranges present in this tile.
    int w0 = si, w1 = ei;
#pragma unroll
    for (int off = 16; off >= 1; off >>= 1) {
        w0 = min(w0, __shfl_xor(w0, off, 32));
        w1 = max(w1, __shfl_xor(w1, off, 32));
    }

    // Register-resident sorted top-K (ascending d^2, ties keep lower index).
    float bd[K_NEI];
    int   bj[K_NEI];
#pragma unroll
    for (int k = 0; k < K_NEI; ++k) { bd[k] = __builtin_inff(); bj[k] = i; }

    for (int j0 = w0; j0 < w1; j0 += TILE_M) {
        // Every lane's candidate column is j0 + (lane & 15).
        const int c  = j0 + l16;
        const int cc = min(c, n - 1);
        const float cx = pos[3 * cc + 0];
        const float cy = pos[3 * cc + 1];
        const float cz = pos[3 * cc + 2];
        const float normC = cx * cx + cy * cy + cz * cz;

        v2f bfrag;
        if (half == 0) { bfrag.x = cx; bfrag.y = cy; }
        else           { bfrag.x = cz; bfrag.y = 0.0f; }

        v8f cacc;
#pragma unroll
        for (int v = 0; v < 8; ++v) cacc[v] = normT_reg[v] + normC;

        // D = (-2*Pt) * Pc^T + (|Pt|^2 + |Pc|^2)  ==  squared distances.
        // 8 args: (neg_a, A, neg_b, B, c_mod, C, reuse_a, reuse_b)
        v8f d = __builtin_amdgcn_wmma_f32_16x16x4_f32(
            false, afrag, false, bfrag, (short)0, cacc, false, false);

        __syncthreads();
#pragma unroll
        for (int v = 0; v < 8; ++v) ldsD2[v + 8 * half][l16] = d[v];
        __syncthreads();

        // Lanes 0-15: each owns one target row; masked insertion into top-K.
        if (half == 0 && i < n) {
            const int jmax = min(TILE_M, w1 - j0);
            for (int t = 0; t < jmax; ++t) {
                const int   j  = j0 + t;
                const float d2 = ldsD2[l16][t];
                // same molecule <=> j in [si,ei) because batch is sorted.
                const bool ok = (j >= si) & (j < ei) & (j != i) & (d2 <= CUTOFF2);
                if (ok && d2 < bd[K_NEI - 1]) {
                    bd[K_NEI - 1] = d2;
                    bj[K_NEI - 1] = j;
#pragma unroll
                    for (int k = K_NEI - 1; k > 0; --k) {
                        if (bd[k] < bd[k - 1]) {
                            float td = bd[k]; bd[k] = bd[k - 1]; bd[k - 1] = td;
                            int   tj = bj[k]; bj[k] = bj[k - 1]; bj[k - 1] = tj;
                        }
                    }
                }
            }
        }
    }

    // Emit edges: out = [rows (N*K) | cols (N*K) | weights (N*K)] as f32.
    if (half == 0 && i < n) {
        const long long NK   = (long long)n * K_NEI;
        const long long base = (long long)i * K_NEI;
        float* rowOut = out;
        float* colOut = out + NK;
        float* wOut   = out + 2 * NK;
#pragma unroll
        for (int k = 0; k < K_NEI; ++k) {
            const bool valid = bd[k] < __builtin_inff();
            rowOut[base + k] = (float)i;
            colOut[base + k] = (float)(valid ? bj[k] : i);
            wOut[base + k]   = valid ? sqrtf(bd[k]) : 0.0f;
        }
    }
}

// ---------------------------------------------------------------------------
// Launch
// ---------------------------------------------------------------------------
extern "C" void kernel_launch(void* const* d_in, const int* in_sizes, int n_in,
                              void* d_out, int out_size, void* d_ws, size_t ws_size,
                              hipStream_t stream) {
    const float* pos   = (const float*)d_in[0];   // [N,3] f32
    const int*   batch = (const int*)d_in[1];     // [N]   i32 (sorted)
    const int    N     = in_sizes[1];

    // Workspace: molecule start/end bounds (batch values < N always).
    int* mstart = (int*)d_ws;
    int* mend   = mstart + N;   // generous upper bound on #molecules

    mol_bounds_kernel<<<(N + 255) / 256, 256, 0, stream>>>(batch, N, mstart, mend);

    const int tiles = (N + TILE_M - 1) / TILE_M;
    radius_graph_wmma_kernel<<<tiles, 32, 0, stream>>>(
        pos, batch, mstart, mend, (float*)d_out, N);
}